// QuantumCoherentNeuron_4174708211977
// MI455X (gfx1250) — compile-verified
//
#include <hip/hip_runtime.h>
#include <math.h>

// ---------------- problem constants ----------------
constexpr int BATCH = 16;
constexpr int T     = 64;
constexpr int NN    = 4096;               // neurons
constexpr int NB    = NN / 16;            // 256 16-wide blocks
constexpr float ENT_S     = 0.6f;
constexpr float SUP_DECAY = 0.95f;
constexpr float COHER     = 0.85f;
constexpr float LR_ADAPT  = 0.001f;
// exp(-1/15), exp(-1/5)
#define MEM_DECAY 0.93550699f
#define SYN_DECAY 0.81873075f
// ENT_S / BATCH  (fold the 1/B of corr into the EMA blend)
#define CORR_SCALE (ENT_S / 16.0f)        // 0.0375

typedef float v2f __attribute__((ext_vector_type(2)));
typedef float v8f __attribute__((ext_vector_type(8)));

// ---------------------------------------------------------------------------
// Per-neuron state transition (shared by fused step kernel and final kernel)
// ---------------------------------------------------------------------------
struct StepOut { float V, a0, a1, at, spk; };

__device__ __forceinline__ StepOut state_math(float Vp, float a0p, float a1p,
                                              float atp, float entRaw, float xb,
                                              float syn, float nz, float qu) {
  StepOut o;
  float V = Vp * MEM_DECAY + (xb + 0.1f * entRaw) * SYN_DECAY + syn * 0.1f;
  const float ph = 0.1f * V;
  float a0n = a0p * __cosf(ph) * SUP_DECAY;
  float a1n = a1p * __sinf(ph) * SUP_DECAY;
  const float inv = rsqrtf(a0n * a0n + a1n * a1n + 1e-8f);
  a0n *= inv; a1n *= inv;
  const float thr = atp + qu * nz;
  const float sig = 1.0f / (1.0f + __expf(-thr));
  const float spk = (a1n * a1n > sig) ? 1.0f : 0.0f;
  V = (spk > 0.5f) ? V * 0.1f : V;
  o.V = V; o.a0 = a0n; o.a1 = a1n; o.spk = spk;
  o.at = fminf(fmaxf(atp + LR_ADAPT * (spk - 0.1f), 0.1f), 2.0f);
  return o;
}

// ===========================================================================
// Kernel 0: (re)initialize persistent state (parity-0 buffers) every call
// ===========================================================================
__global__ __launch_bounds__(256) void qcn_init(float* __restrict__ E,
                                                float* __restrict__ V2,
                                                float* __restrict__ a02,
                                                float* __restrict__ a12,
                                                float* __restrict__ at2,
                                                const float* __restrict__ thr0) {
  size_t idx = (size_t)blockIdx.x * 256 + threadIdx.x;
  if (idx < (size_t)NN * NN) E[idx] = 0.0f;
  if (idx < NN) {
    V2[idx]  = 0.0f;
    a02[idx] = sqrtf(1.0f - COHER);
    a12[idx] = sqrtf(COHER);
    at2[idx] = thr0[idx];
  }
}

// ===========================================================================
// Kernel 1: Xbar[t,n] = mean_b input[b,t,n]
// ===========================================================================
__global__ __launch_bounds__(256) void qcn_xbar(const float* __restrict__ IN,
                                                float* __restrict__ Xbar) {
  int idx = blockIdx.x * 256 + threadIdx.x;        // t*NN + n
  if (idx >= T * NN) return;
  float s = 0.0f;
#pragma unroll
  for (int b = 0; b < BATCH; ++b) s += IN[(size_t)b * T * NN + idx];
  Xbar[idx] = s * (1.0f / 16.0f);
}

// ===========================================================================
// Kernel 2: SynMean = Xbar @ W   (one pass over W, WMMA f32 16x16x4)
//   (x @ W).mean(0) == mean(x) @ W  -> hoisted out of the time loop entirely;
//   reads the 64 MB W exactly once instead of 64 times (saves ~4 GB traffic).
// ===========================================================================
__global__ __launch_bounds__(256) void qcn_synmean(const float* __restrict__ Xbar,
                                                   const float* __restrict__ W,
                                                   float* __restrict__ SynMean) {
  const int mb   = blockIdx.y;                     // 0..3   (M = T = 64)
  const int wave = threadIdx.x >> 5;
  const int lane = threadIdx.x & 31;
  const int half = lane >> 4;
  const int l    = lane & 15;
  const int nb   = blockIdx.x * 8 + wave;          // 0..255
  const int ncol = nb * 16 + l;

  v8f c = {};
  for (int k0 = 0; k0 < NN; k0 += 4) {
    const int kk = k0 + 2 * half;                  // K index held by this lane half
    v2f a, b;
    a.x = Xbar[(size_t)(mb * 16 + l) * NN + kk];       // A[M=l, K=kk]
    a.y = Xbar[(size_t)(mb * 16 + l) * NN + kk + 1];   // A[M=l, K=kk+1]
    b.x = W[(size_t)kk * NN + ncol];                   // B[K=kk,   N=ncol]
    b.y = W[(size_t)(kk + 1) * NN + ncol];             // B[K=kk+1, N=ncol]
    c = __builtin_amdgcn_wmma_f32_16x16x4_f32(false, a, false, b,
                                              (short)0, c, false, false);
  }
#pragma unroll
  for (int r = 0; r < 8; ++r)
    SynMean[(size_t)(mb * 16 + r + 8 * half) * NN + ncol] = c[r];
}

// ===========================================================================
// Kernel 3 (per step t): FUSED
//   Phase 0: state update for step t-1 (every WG recomputes all 4096 neurons
//            -> new V cached in LDS; only the owner WG of each 16-neuron strip
//            persists state (ping-pong parity) and streams outputs via
//            non-temporal stores so the 67 MB of outputs never evict E from
//            the 192 MB L2).
//   Phase 1: E <- 0.0375*(X^T X) + 0.4*E ; ent = E @ V   (V read from LDS)
//            corr tile = 4x chained V_WMMA_F32_16X16X4_F32 (K = batch = 16),
//            EMA blend + E*V partial fused in the C registers.
//   The kernel boundary between consecutive steps provides the grid-wide
//   sync between ent(t-1) production and the state update that consumes it.
// ===========================================================================
__global__ __launch_bounds__(256) void qcn_step(const float* __restrict__ IN,
                                                float* __restrict__ E,
                                                const float* __restrict__ Xbar,
                                                const float* __restrict__ SynMean,
                                                const float* __restrict__ noise,
                                                const float* __restrict__ qu,
                                                float* __restrict__ V2,
                                                float* __restrict__ a02,
                                                float* __restrict__ a12,
                                                float* __restrict__ at2,
                                                float* __restrict__ ent2,
                                                float* __restrict__ out,
                                                int t) {
  const int ib   = blockIdx.x;                     // 0..255 (owns rows ib*16..+15)
  const int tid  = threadIdx.x;
  const int wave = tid >> 5;
  const int lane = tid & 31;
  const int half = lane >> 4;
  const int l    = lane & 15;

  __shared__ float ldsV[NN];                       // V entering step t (16 KB)
  __shared__ float part[8][16];

  const int pc = t & 1;                            // parity of state entering t
  const int pp = pc ^ 1;                           // parity of state entering t-1

  // ---- Phase 0: compute state entering step t ----
  if (t == 0) {
    for (int i = tid; i < NN; i += 256) ldsV[i] = 0.0f;   // V0 = 0
  } else {
    const float* Vp  = V2  + (size_t)pp * NN;
    const float* a0p = a02 + (size_t)pp * NN;
    const float* a1p = a12 + (size_t)pp * NN;
    const float* atp = at2 + (size_t)pp * NN;
    const float* ep  = ent2 + (size_t)pp * NN;
    float* Vn  = V2  + (size_t)pc * NN;
    float* a0n = a02 + (size_t)pc * NN;
    float* a1n = a12 + (size_t)pc * NN;
    float* atn = at2 + (size_t)pc * NN;
    const int tm1 = t - 1;
    const size_t row = (size_t)tm1 * NN;
    const size_t BTN = (size_t)BATCH * T * NN;
    const int own_lo = ib * 16;

    for (int i = tid; i < NN; i += 256) {
      StepOut s = state_math(Vp[i], a0p[i], a1p[i], atp[i], ep[i],
                             Xbar[row + i], SynMean[row + i], noise[row + i],
                             qu[i]);
      ldsV[i] = s.V;
      if ((unsigned)(i - own_lo) < 16u) {          // owner WG persists + outputs
        Vn[i] = s.V; a0n[i] = s.a0; a1n[i] = s.a1; atn[i] = s.at;
        const size_t base = row + i;
#pragma unroll
        for (int b = 0; b < BATCH; ++b) {
          const size_t o = (size_t)b * T * NN + base;
          __builtin_nontemporal_store(s.spk, &out[o]);
          __builtin_nontemporal_store(s.V,   &out[BTN + o]);
          __builtin_nontemporal_store(s.a0,  &out[2 * BTN + 2 * o]);
          __builtin_nontemporal_store(s.a1,  &out[2 * BTN + 2 * o + 1]);
        }
      }
    }
  }
  __syncthreads();

  // ---- Phase 1: E update + ent = E @ V ----
  const float* x = IN + (size_t)t * NN;            // x[b][n] = x[b*T*NN + n]
  const size_t bs = (size_t)T * NN;

  v2f afrag[4];
  const int icol = ib * 16 + l;
#pragma unroll
  for (int s = 0; s < 4; ++s) {
    const int b0 = 4 * s + 2 * half;
    afrag[s].x = x[(size_t)b0 * bs + icol];
    afrag[s].y = x[(size_t)(b0 + 1) * bs + icol];
  }

  float acc[8];
#pragma unroll
  for (int r = 0; r < 8; ++r) acc[r] = 0.0f;

  for (int jb = wave; jb < NB; jb += 8) {
    const int jcol = jb * 16 + l;
    v8f c = {};
#pragma unroll
    for (int s = 0; s < 4; ++s) {
      const int b0 = 4 * s + 2 * half;
      v2f bfrag;
      bfrag.x = x[(size_t)b0 * bs + jcol];
      bfrag.y = x[(size_t)(b0 + 1) * bs + jcol];
      c = __builtin_amdgcn_wmma_f32_16x16x4_f32(false, afrag[s], false, bfrag,
                                                (short)0, c, false, false);
    }
    const float vseg = ldsV[jcol];
#pragma unroll
    for (int r = 0; r < 8; ++r) {
      const size_t eidx = (size_t)(ib * 16 + r + 8 * half) * NN + jcol;
      const float en = CORR_SCALE * c[r] + (1.0f - ENT_S) * E[eidx];
      E[eidx] = en;
      acc[r] += en * vseg;                         // E @ V partial (column l)
    }
  }

  // reduce over the 16 lanes of each half-wave (columns within a tile)
#pragma unroll
  for (int r = 0; r < 8; ++r) {
#pragma unroll
    for (int off = 8; off >= 1; off >>= 1)
      acc[r] += __shfl_xor(acc[r], off, 16);
  }

  if (l == 0) {
#pragma unroll
    for (int r = 0; r < 8; ++r) part[wave][r + 8 * half] = acc[r];
  }
  __syncthreads();
  if (tid < 16) {
    float s = 0.0f;
#pragma unroll
    for (int w = 0; w < 8; ++w) s += part[w][tid];
    ent2[(size_t)pc * NN + ib * 16 + tid] = s;     // this WG owns these rows
  }
}

// ===========================================================================
// Kernel 4: final state update (step T-1) -> outputs only
// ===========================================================================
__global__ __launch_bounds__(256) void qcn_final(const float* __restrict__ Xbar,
                                                 const float* __restrict__ SynMean,
                                                 const float* __restrict__ noise,
                                                 const float* __restrict__ qu,
                                                 const float* __restrict__ V2,
                                                 const float* __restrict__ a02,
                                                 const float* __restrict__ a12,
                                                 const float* __restrict__ at2,
                                                 const float* __restrict__ ent2,
                                                 float* __restrict__ out) {
  const int i = blockIdx.x * 256 + threadIdx.x;
  if (i >= NN) return;
  const int t = T - 1;
  const int p = t & 1;                             // parity of state entering t
  const size_t row = (size_t)t * NN;
  StepOut s = state_math(V2[(size_t)p * NN + i], a02[(size_t)p * NN + i],
                         a12[(size_t)p * NN + i], at2[(size_t)p * NN + i],
                         ent2[(size_t)p * NN + i],
                         Xbar[row + i], SynMean[row + i], noise[row + i], qu[i]);
  const size_t BTN = (size_t)BATCH * T * NN;
  const size_t base = row + i;
#pragma unroll
  for (int b = 0; b < BATCH; ++b) {
    const size_t o = (size_t)b * T * NN + base;
    __builtin_nontemporal_store(s.spk, &out[o]);
    __builtin_nontemporal_store(s.V,   &out[BTN + o]);
    __builtin_nontemporal_store(s.a0,  &out[2 * BTN + 2 * o]);
    __builtin_nontemporal_store(s.a1,  &out[2 * BTN + 2 * o + 1]);
  }
}

// ===========================================================================
extern "C" void kernel_launch(void* const* d_in, const int* in_sizes, int n_in,
                              void* d_out, int out_size, void* d_ws, size_t ws_size,
                              hipStream_t stream) {
  const float* IN    = (const float*)d_in[0];   // input_current  [B,T,N]
  const float* W     = (const float*)d_in[1];   // synaptic_weights [N,N]
  const float* thr0  = (const float*)d_in[2];   // adaptive_threshold [N]
  const float* qu    = (const float*)d_in[3];   // quantum_uncertainty [N]
  const float* noise = (const float*)d_in[4];   // threshold_noise [T,N]
  float* out = (float*)d_out;

  // workspace layout (~68.8 MB):
  //   E | Xbar | SynMean | ent[2] | V[2] | a0[2] | a1[2] | at[2]
  float* E       = (float*)d_ws;
  float* Xbar    = E + (size_t)NN * NN;
  float* SynMean = Xbar + (size_t)T * NN;
  float* ent2    = SynMean + (size_t)T * NN;
  float* V2      = ent2 + 2 * NN;
  float* a02     = V2 + 2 * NN;
  float* a12     = a02 + 2 * NN;
  float* at2     = a12 + 2 * NN;

  qcn_init<<<(int)(((size_t)NN * NN + 255) / 256), 256, 0, stream>>>(
      E, V2, a02, a12, at2, thr0);
  qcn_xbar<<<(T * NN + 255) / 256, 256, 0, stream>>>(IN, Xbar);
  qcn_synmean<<<dim3(32, 4), 256, 0, stream>>>(Xbar, W, SynMean);

  for (int t = 0; t < T; ++t) {
    qcn_step<<<NB, 256, 0, stream>>>(IN, E, Xbar, SynMean, noise, qu,
                                     V2, a02, a12, at2, ent2, out, t);
  }
  qcn_final<<<NN / 256, 256, 0, stream>>>(Xbar, SynMean, noise, qu,
                                          V2, a02, a12, at2, ent2, out);
}